// KoLeoLossData_43946105373284
// MI455X (gfx1250) — compile-verified
//
#include <hip/hip_runtime.h>

typedef __attribute__((ext_vector_type(2))) float v2f;
typedef __attribute__((ext_vector_type(8))) float v8f;

#define BATCH 8
#define T 4096
#define D 256
#define NT (T / 16)          // 256 column tiles
#define ROWS_PER_WG 64
#define LDS_STRIDE 260       // 256 + 4 : stride == 4 (mod 64) -> conflict-free WMMA fragment ds reads

// ---------------------------------------------------------------------------
// Async-stage one 16x256 f32 tile (16 KB) from global into LDS using the
// CDNA5 direct memory->LDS path (GLOBAL_LOAD_ASYNC_TO_LDS_B128, ASYNCcnt).
// 128 threads x 8 b128 ops; LDS rows padded to LDS_STRIDE dwords.
// ---------------------------------------------------------------------------
__device__ __forceinline__ void stage_tile_async(const float* gbase,
                                                 unsigned ldsByteBase, int tid)
{
#pragma unroll
    for (int i = 0; i < 8; ++i) {
        const int f   = tid + i * 128;          // float4 index 0..1023
        const int row = f >> 6;
        const int c4  = f & 63;
        unsigned lAddr = ldsByteBase + (unsigned)(row * (LDS_STRIDE * 4) + c4 * 16);
        unsigned gOff  = (unsigned)f * 16u;     // byte offset within tile
        asm volatile("global_load_async_to_lds_b128 %0, %1, %2"
                     :
                     : "v"(lAddr), "v"(gOff), "s"(gbase)
                     : "memory");
    }
}

// ---------------------------------------------------------------------------
// Kernel 1: fused Gram-matrix (X X^T) + row argmax (diag forced to -1.0).
// 512 WGs = 8 batches x 64 row-strips; 4 waves/WG, each wave owns a 16-row
// strip. A (64x256) staged in LDS once; B tiles double-buffered via async
// loads. 64 x v_wmma_f32_16x16x4_f32 per 16x16 output tile (2 acc chains).
// ---------------------------------------------------------------------------
extern "C" __global__ __launch_bounds__(128)
void koleo_nn_argmax(const float* __restrict__ x, int* __restrict__ nn)
{
    extern __shared__ float lds[];
    float* As  = lds;                                      // 64 x LDS_STRIDE
    float* Bs0 = lds + ROWS_PER_WG * LDS_STRIDE;           // 16 x LDS_STRIDE
    float* Bs1 = Bs0 + 16 * LDS_STRIDE;                    // 16 x LDS_STRIDE

    const int tid    = threadIdx.x;
    const int lane   = tid & 31;
    const int w      = tid >> 5;
    const int laneLo = lane & 15;
    const int laneHi = lane >> 4;

    const int wg      = blockIdx.x;
    const int b       = wg >> 6;            // 64 strips per batch
    const int strip   = wg & 63;
    const int rowBase = strip * ROWS_PER_WG;

    const float* xb = x + (size_t)b * T * D;

    // LDS byte offsets (generic pointer low 32 bits == LDS offset per flat-aperture rule)
    const unsigned bs0Byte = (unsigned)(uintptr_t)Bs0;
    const unsigned bs1Byte = (unsigned)(uintptr_t)Bs1;

    // ---- stage A strip (64 rows x 256 f32) into LDS, float4 coalesced ----
    {
        const float4* src = reinterpret_cast<const float4*>(xb + (size_t)rowBase * D);
        for (int i = tid; i < ROWS_PER_WG * (D / 4); i += 128) {
            int r = i >> 6;          // row
            int c = i & 63;          // float4 index within row
            float4 v = src[r * (D / 4) + c];
            reinterpret_cast<float4*>(&As[r * LDS_STRIDE + c * 4])[0] = v;
        }
    }

    // ---- prologue: async-stage B tile 0 into buffer 0 ----
    stage_tile_async(xb, bs0Byte, tid);

    float best[8];
    int   bidx[8];
#pragma unroll
    for (int r = 0; r < 8; ++r) { best[r] = -3.402823466e38f; bidx[r] = 0; }

    // C/D layout: VGPR r holds M=r (lanes 0-15) and M=r+8 (lanes 16-31), N=laneLo
    const int myRow0 = rowBase + w * 16 + laneHi * 8;

    const float* aBase  = &As [(w * 16 + laneLo) * LDS_STRIDE + 2 * laneHi];
    const float* bBase0 = &Bs0[laneLo * LDS_STRIDE + 2 * laneHi];
    const float* bBase1 = &Bs1[laneLo * LDS_STRIDE + 2 * laneHi];

    for (int jt = 0; jt < NT; ++jt) {
        const int cur = jt & 1;

        // all waves finished reading the buffer we are about to overwrite
        __syncthreads();

        if (jt + 1 < NT) {
            stage_tile_async(xb + (size_t)(jt + 1) * 16 * D,
                             cur ? bs0Byte : bs1Byte, tid);
            // in-order completion: <=8 outstanding => current tile's 8 are done
            asm volatile("s_wait_asynccnt 8" ::: "memory");
        } else {
            asm volatile("s_wait_asynccnt 0" ::: "memory");
        }
        __syncthreads();   // every wave's portion of the current tile is visible

        // ---- 16x16 fp32 tile: K=256 in steps of 4; two accumulation chains ----
        const float* bBase = cur ? bBase1 : bBase0;
        v8f acc0 = {}, acc1 = {};
#pragma unroll 4
        for (int kk = 0; kk < D / 4; kk += 2) {
            // A 16x4 layout: lane m<16 -> K=0,1 ; lane m+16 -> K=2,3  (float2 each)
            v2f a0 = *reinterpret_cast<const v2f*>(aBase + kk * 4);
            v2f b0 = *reinterpret_cast<const v2f*>(bBase + kk * 4);
            v2f a1 = *reinterpret_cast<const v2f*>(aBase + kk * 4 + 4);
            v2f b1 = *reinterpret_cast<const v2f*>(bBase + kk * 4 + 4);
            acc0 = __builtin_amdgcn_wmma_f32_16x16x4_f32(false, a0, false, b0,
                                                         (short)0, acc0, false, false);
            acc1 = __builtin_amdgcn_wmma_f32_16x16x4_f32(false, a1, false, b1,
                                                         (short)0, acc1, false, false);
        }
        v8f acc = acc0 + acc1;

        // ---- fused argmax epilogue (diag = -1.0 like reference) ----
        const int colG = jt * 16 + laneLo;
#pragma unroll
        for (int r = 0; r < 8; ++r) {
            float v = acc[r];
            if (colG == myRow0 + r) v = -1.0f;
            if (v > best[r]) { best[r] = v; bidx[r] = colG; }
        }
    }

    // ---- cross-lane argmax reduction within each 16-lane half ----
#pragma unroll
    for (int r = 0; r < 8; ++r) {
        float bv = best[r];
        int   bi = bidx[r];
#pragma unroll
        for (int m = 8; m >= 1; m >>= 1) {
            float ov = __shfl_xor(bv, m, 32);
            int   oi = __shfl_xor(bi, m, 32);
            // tie-break on lower index to match jnp.argmax first-occurrence
            if (ov > bv || (ov == bv && oi < bi)) { bv = ov; bi = oi; }
        }
        best[r] = bv; bidx[r] = bi;
    }

    if (laneLo == 0) {
#pragma unroll
        for (int r = 0; r < 8; ++r)
            nn[(size_t)b * T + myRow0 + r] = bidx[r];
    }
}

// ---------------------------------------------------------------------------
// Kernel 2: one wave per (b,t): partial[u] = -log(||x_t - x_nn + 1e-8|| + 1e-8)
// ---------------------------------------------------------------------------
extern "C" __global__ __launch_bounds__(256)
void koleo_dist_log(const float* __restrict__ x, const int* __restrict__ nn,
                    float* __restrict__ partial)
{
    const int lane = threadIdx.x & 31;
    const int u    = blockIdx.x * 8 + (threadIdx.x >> 5);   // [0, BATCH*T)
    const int b    = u >> 12;                               // u / T
    const int nnI  = nn[u];

    const float4* xr = reinterpret_cast<const float4*>(x + (size_t)u * D);
    const float4* xn = reinterpret_cast<const float4*>(x + ((size_t)b * T + nnI) * D);

    float s = 0.f;
#pragma unroll
    for (int c = 0; c < 2; ++c) {
        float4 a = xr[lane + c * 32];
        float4 q = xn[lane + c * 32];
        float d0 = a.x - q.x + 1e-8f;
        float d1 = a.y - q.y + 1e-8f;
        float d2 = a.z - q.z + 1e-8f;
        float d3 = a.w - q.w + 1e-8f;
        s += d0 * d0 + d1 * d1 + d2 * d2 + d3 * d3;
    }
#pragma unroll
    for (int m = 16; m >= 1; m >>= 1)
        s += __shfl_xor(s, m, 32);

    if (lane == 0)
        partial[u] = -__logf(sqrtf(s) + 1e-8f);
}

// ---------------------------------------------------------------------------
// Kernel 3: deterministic single-block reduction -> mean -> d_out[0]
// ---------------------------------------------------------------------------
extern "C" __global__ __launch_bounds__(1024)
void koleo_reduce(const float* __restrict__ partial, float* __restrict__ out)
{
    __shared__ float sm[1024];
    float s = 0.f;
    for (int i = threadIdx.x; i < BATCH * T; i += 1024)
        s += partial[i];
    sm[threadIdx.x] = s;
    __syncthreads();
    for (int off = 512; off > 0; off >>= 1) {
        if (threadIdx.x < off) sm[threadIdx.x] += sm[threadIdx.x + off];
        __syncthreads();
    }
    if (threadIdx.x == 0)
        out[0] = sm[0] * (1.0f / (float)(BATCH * T));
}

// ---------------------------------------------------------------------------
extern "C" void kernel_launch(void* const* d_in, const int* in_sizes, int n_in,
                              void* d_out, int out_size, void* d_ws, size_t ws_size,
                              hipStream_t stream)
{
    const float* x = (const float*)d_in[0];

    int*   nn      = (int*)d_ws;                                        // B*T ints   (128 KB)
    float* partial = (float*)((char*)d_ws + (size_t)BATCH * T * 4);     // B*T floats (128 KB)

    const size_t ldsBytes = (size_t)(ROWS_PER_WG + 32) * LDS_STRIDE * sizeof(float); // 99840

    koleo_nn_argmax<<<BATCH * (T / ROWS_PER_WG), 128, ldsBytes, stream>>>(x, nn);
    koleo_dist_log<<<(BATCH * T) / 8, 256, 0, stream>>>(x, nn, partial);
    koleo_reduce<<<1, 1024, 0, stream>>>(partial, (float*)d_out);
}